// LinearAttention_48902497632777
// MI455X (gfx1250) — compile-verified
//
#include <hip/hip_runtime.h>
#include <math.h>

// ---------------------------------------------------------------------------
// Linear attention forward for MI455X (gfx1250), f32 end-to-end via
// V_WMMA_F32_16X16X4_F32, with async global->LDS double buffering.
// B=4, T=4096, D=1024, H=16, HD=64.
// ---------------------------------------------------------------------------

typedef __attribute__((ext_vector_type(2))) float v2f;
typedef __attribute__((ext_vector_type(8))) float v8f;

#define DD    1024      // model dim
#define TSEQ  4096      // sequence length
#define NB    4         // batch
#define NH    16        // heads
#define HDIM  64        // head dim
#define BTROWS (NB*TSEQ) // 16384

static __device__ __forceinline__ v8f wmma_f32(v2f a, v2f b, v8f c) {
  // 8-arg pattern: (neg_a, A, neg_b, B, c_mod, C, reuse_a, reuse_b)
  return __builtin_amdgcn_wmma_f32_16x16x4_f32(
      false, a, false, b, (short)0, c, false, false);
}

// phi(x) = elu(x) + 1 = x+1 (x>0) else expm1(x)+1
static __device__ __forceinline__ float phi(float v) {
  return (v > 0.0f) ? (v + 1.0f) : (expm1f(v) + 1.0f);
}

// ---------------------------------------------------------------------------
// Async global -> LDS copy (B128) + ASYNCcnt wait.
// Builtin param 0 is AS1 (global) int4*, param 1 is AS3 (LDS) int4*
// (from the round-2 diagnostic: 'int __vector_size__(16) __device__ *').
// ---------------------------------------------------------------------------
#if __has_builtin(__builtin_amdgcn_global_load_async_to_lds_b128)
#define HAVE_ASYNC_B128 1
#else
#define HAVE_ASYNC_B128 0
#endif
#if __has_builtin(__builtin_amdgcn_s_wait_asynccnt)
#define HAVE_WAIT_ASYNC 1
#else
#define HAVE_WAIT_ASYNC 0
#endif

typedef __attribute__((__vector_size__(16))) int i32x4;
typedef __attribute__((address_space(1))) i32x4  g_i32x4;   // global
typedef __attribute__((address_space(3))) i32x4  l_i32x4;   // LDS
typedef __attribute__((address_space(3))) float  lds_f;

static __device__ __forceinline__ void async_b128(const float* g, float* l) {
#if HAVE_ASYNC_B128
  __builtin_amdgcn_global_load_async_to_lds_b128(
      (g_i32x4*)g, (l_i32x4*)l, 0, 0);
#else
  unsigned loff = (unsigned)(size_t)(lds_f*)l;   // LDS byte offset
  asm volatile("global_load_async_to_lds_b128 %0, %1, off"
               :: "v"(loff), "v"(g) : "memory");
#endif
}

template <int N>
static __device__ __forceinline__ void wait_async() {
#if HAVE_WAIT_ASYNC
  __builtin_amdgcn_s_wait_asynccnt(N);
#else
  asm volatile("s_wait_asynccnt %0" :: "i"(N) : "memory");
#endif
}

// ---------------------------------------------------------------------------
// GEMM:  out[M=16384, N=1024] = epilogue( X[M,1024] @ W[N,1024]^T )
// mode bit0: apply phi; mode bit1: zero rows where mask[row] is true.
// Block: 128 threads = 4 waves; block tile 128x128; wave tile 64x64.
// K staged via async-to-LDS, double buffered (2 x 2 x 18KB = 72KB LDS).
// ---------------------------------------------------------------------------
constexpr int BMT  = 128;
constexpr int BNT  = 128;
constexpr int KC   = 32;
constexpr int LSTR = KC + 4;     // 36 floats: 16B-aligned rows, conflict-free
constexpr int NCHUNK = DD / KC;  // 32

__global__ __launch_bounds__(128) void gemm_phi_mask(
    const float* __restrict__ X, const float* __restrict__ W,
    float* __restrict__ out, const unsigned char* __restrict__ mask, int mode)
{
  __shared__ float Xs[2][BMT * LSTR];
  __shared__ float Ws[2][BNT * LSTR];

  const int tid  = threadIdx.x;
  const int lane = tid & 31;
  const int wv   = tid >> 5;
  const int wy   = wv >> 1, wx = wv & 1;
  const int lo   = lane & 15, hi = lane >> 4;
  const int m0   = blockIdx.x * BMT;
  const int n0   = blockIdx.y * BNT;

  // issue one KC-wide chunk: 8 B128 per thread per matrix (16 asyncs total)
  auto issue_chunk = [&](int k0, int buf) {
#pragma unroll
    for (int it = 0; it < 8; ++it) {
      int idx = it * 128 + tid;          // float4 index within tile
      int row = idx >> 3;                // 8 float4 per row
      int c4  = (idx & 7) << 2;
      async_b128(X + (size_t)(m0 + row) * DD + k0 + c4,
                 &Xs[buf][row * LSTR + c4]);
      async_b128(W + (size_t)(n0 + row) * DD + k0 + c4,
                 &Ws[buf][row * LSTR + c4]);
    }
  };

  v8f acc[4][4] = {};

  issue_chunk(0, 0);
  for (int ch = 0; ch < NCHUNK; ++ch) {
    const int buf = ch & 1;
    if (ch + 1 < NCHUNK) {
      issue_chunk((ch + 1) * KC, buf ^ 1);  // prefetch next chunk
      wait_async<16>();                     // cur chunk's 16 asyncs retired
    } else {
      wait_async<0>();
    }
    __syncthreads();                        // cur buffer published block-wide

    const float* xs = &Xs[buf][0];
    const float* wsh = &Ws[buf][0];
#pragma unroll
    for (int kk = 0; kk < KC; kk += 4) {
      const int kb = kk + 2 * hi;           // lanes 16-31 hold K=2,3
      v2f af[4], bf[4];
#pragma unroll
      for (int mi = 0; mi < 4; ++mi)
        af[mi] = *(const v2f*)&xs[(wy * 64 + mi * 16 + lo) * LSTR + kb];
#pragma unroll
      for (int ni = 0; ni < 4; ++ni)
        bf[ni] = *(const v2f*)&wsh[(wx * 64 + ni * 16 + lo) * LSTR + kb];
#pragma unroll
      for (int mi = 0; mi < 4; ++mi)
#pragma unroll
        for (int ni = 0; ni < 4; ++ni)
          acc[mi][ni] = wmma_f32(af[mi], bf[ni], acc[mi][ni]);
    }
    __syncthreads();                        // done reading buf before reuse
  }

  // epilogue: C layout -> row = base + r + 8*hi, col = base + lo
#pragma unroll
  for (int mi = 0; mi < 4; ++mi)
#pragma unroll
    for (int ni = 0; ni < 4; ++ni)
#pragma unroll
      for (int r = 0; r < 8; ++r) {
        int row = m0 + wy * 64 + mi * 16 + r + 8 * hi;
        int col = n0 + wx * 64 + ni * 16 + lo;
        float v = acc[mi][ni][r];
        if (mode & 1) v = phi(v);
        if ((mode & 2) && mask[row]) v = 0.0f;   // mask flat index == bt row
        out[(size_t)row * DD + col] = v;
      }
}

// ---------------------------------------------------------------------------
// kv[b,h] += K_h^T V_h  (64x64 per head), T split 8 blocks x 4 waves,
// f32 atomics into pre-zeroed kv.
// ---------------------------------------------------------------------------
__global__ __launch_bounds__(128) void kv_accum(
    const float* __restrict__ Kb, const float* __restrict__ Vb,
    float* __restrict__ kv)
{
  const int bh = blockIdx.x;            // 0..63
  const int b = bh >> 4, h = bh & 15;
  const int tid = threadIdx.x, lane = tid & 31, wv = tid >> 5;
  const int lo = lane & 15, hi = lane >> 4;
  const int t0 = blockIdx.y * 512 + wv * 128;   // 128 t-rows per wave

  const float* Kp = Kb + (size_t)b * TSEQ * DD + (size_t)h * HDIM;
  const float* Vp = Vb + (size_t)b * TSEQ * DD + (size_t)h * HDIM;

  v8f acc[4][4] = {};
  for (int kt = 0; kt < 128; kt += 4) {
    const int tr = t0 + kt + 2 * hi;
    v2f af[4], bf[4];
#pragma unroll
    for (int mi = 0; mi < 4; ++mi) {    // A[m=d][k=t] = K[t][d]
      int col = mi * 16 + lo;
      af[mi].x = Kp[(size_t)tr * DD + col];
      af[mi].y = Kp[(size_t)(tr + 1) * DD + col];
    }
#pragma unroll
    for (int ni = 0; ni < 4; ++ni) {    // B[k=t][n=e] = V[t][e]
      int col = ni * 16 + lo;
      bf[ni].x = Vp[(size_t)tr * DD + col];
      bf[ni].y = Vp[(size_t)(tr + 1) * DD + col];
    }
#pragma unroll
    for (int mi = 0; mi < 4; ++mi)
#pragma unroll
      for (int ni = 0; ni < 4; ++ni)
        acc[mi][ni] = wmma_f32(af[mi], bf[ni], acc[mi][ni]);
  }

  float* kvp = kv + (size_t)bh * HDIM * HDIM;
#pragma unroll
  for (int mi = 0; mi < 4; ++mi)
#pragma unroll
    for (int ni = 0; ni < 4; ++ni)
#pragma unroll
      for (int r = 0; r < 8; ++r)
        atomicAdd(&kvp[(mi * 16 + r + 8 * hi) * HDIM + ni * 16 + lo],
                  acc[mi][ni][r]);
}

// ---------------------------------------------------------------------------
// ksum[b, h*64+d] += sum_t K[b,t, h*64+d]; T split over blockIdx.y, coalesced.
// ---------------------------------------------------------------------------
__global__ void ksum_kernel(const float* __restrict__ Kb, float* __restrict__ ksum)
{
  const int b   = blockIdx.x >> 2;
  const int col = ((blockIdx.x & 3) << 8) + threadIdx.x;
  const int t0  = blockIdx.y * 256;
  const float* p = Kb + ((size_t)b * TSEQ + t0) * DD + col;
  float s = 0.0f;
  for (int t = 0; t < 256; ++t) s += p[(size_t)t * DD];
  atomicAdd(&ksum[b * DD + col], s);
}

// ---------------------------------------------------------------------------
// norm[bt*H + h] = max(dot(Q[bt, h*64:...], ksum[b, h*64:...]), 1e-6)
// ---------------------------------------------------------------------------
__global__ void norm_kernel(const float* __restrict__ Q,
                            const float* __restrict__ ksum,
                            float* __restrict__ nrm)
{
  const int idx = blockIdx.x * 256 + threadIdx.x;  // bt*NH + h
  const int bt = idx >> 4;
  const int h  = idx & 15;
  const int b  = bt >> 12;                          // bt / TSEQ
  const float* q  = Q + (size_t)bt * DD + h * HDIM;
  const float* ks = ksum + b * DD + h * HDIM;
  float s = 0.0f;
#pragma unroll 8
  for (int i = 0; i < HDIM; ++i) s += q[i] * ks[i];
  nrm[idx] = fmaxf(s, 1e-6f);
}

// ---------------------------------------------------------------------------
// Q[rows, h*64: h*64+64] <- (Q_h @ kv_h) / norm   (in place; each
// (row-block, head) region is owned by exactly one wave).
// Block: 128 threads = 4 waves; wave tile 32 rows x 64 cols; K = 64.
// ---------------------------------------------------------------------------
__global__ __launch_bounds__(128) void qkv_apply(
    float* __restrict__ Q, const float* __restrict__ kv,
    const float* __restrict__ nrm)
{
  const int h  = blockIdx.y;
  const int m0 = blockIdx.x * 128;
  const int b  = m0 >> 12;
  const int tid = threadIdx.x, lane = tid & 31, wv = tid >> 5;
  const int lo = lane & 15, hi = lane >> 4;
  const int wm0 = m0 + wv * 32;

  const float* kvp = kv + (size_t)(b * NH + h) * HDIM * HDIM;

  v8f acc[2][4] = {};
  for (int k0 = 0; k0 < HDIM; k0 += 4) {
    const int kb = k0 + 2 * hi;
    v2f af[2], bf[4];
#pragma unroll
    for (int mi = 0; mi < 2; ++mi)      // A[m=t][k=d] = Q[t][h*64+d]
      af[mi] = *(const v2f*)(Q + (size_t)(wm0 + mi * 16 + lo) * DD + h * HDIM + kb);
#pragma unroll
    for (int ni = 0; ni < 4; ++ni) {    // B[k=d][n=e] = kv[d][e]
      int col = ni * 16 + lo;
      bf[ni].x = kvp[kb * HDIM + col];
      bf[ni].y = kvp[(kb + 1) * HDIM + col];
    }
#pragma unroll
    for (int mi = 0; mi < 2; ++mi)
#pragma unroll
      for (int ni = 0; ni < 4; ++ni)
        acc[mi][ni] = wmma_f32(af[mi], bf[ni], acc[mi][ni]);
  }

#pragma unroll
  for (int mi = 0; mi < 2; ++mi)
#pragma unroll
    for (int ni = 0; ni < 4; ++ni)
#pragma unroll
      for (int r = 0; r < 8; ++r) {
        int row = wm0 + mi * 16 + r + 8 * hi;
        float nv = nrm[row * NH + h];
        Q[(size_t)row * DD + h * HDIM + ni * 16 + lo] = acc[mi][ni][r] / nv;
      }
}

__global__ void zero_f32(float* __restrict__ p, int n)
{
  int i = blockIdx.x * 256 + threadIdx.x;
  if (i < n) p[i] = 0.0f;
}

// ---------------------------------------------------------------------------
// Launch: inputs are x, mask, Wq, Wk, Wv, Wo (setup_inputs order).
// Workspace layout (floats): Q | K | V | kv | ksum | norm  (~203 MB).
// ---------------------------------------------------------------------------
extern "C" void kernel_launch(void* const* d_in, const int* in_sizes, int n_in,
                              void* d_out, int out_size, void* d_ws, size_t ws_size,
                              hipStream_t stream)
{
  (void)in_sizes; (void)n_in; (void)out_size; (void)ws_size;

  const float* x            = (const float*)d_in[0];
  const unsigned char* mask = (const unsigned char*)d_in[1]; // bool, B*T bytes
  const float* Wq           = (const float*)d_in[2];
  const float* Wk           = (const float*)d_in[3];
  const float* Wv           = (const float*)d_in[4];
  const float* Wo           = (const float*)d_in[5];
  float* out                = (float*)d_out;

  float* ws   = (float*)d_ws;
  const size_t slab = (size_t)BTROWS * DD;     // 16.78M floats
  float* Q    = ws;
  float* Kb   = ws + slab;
  float* Vb   = ws + 2 * slab;
  float* kv   = ws + 3 * slab;                 // 64 * 64 * 64 = 262144
  float* ksum = kv + 64 * HDIM * HDIM;         // B * D = 4096
  float* nrm  = ksum + NB * DD;                // BT * H = 262144

  // accumulators must start at zero each call (ws is not re-poisoned)
  const int nz = 64 * HDIM * HDIM + NB * DD;   // kv + ksum (contiguous)
  zero_f32<<<dim3((nz + 255) / 256), dim3(256), 0, stream>>>(kv, nz);

  const dim3 gg(BTROWS / BMT, DD / BNT);       // 128 x 8
  gemm_phi_mask<<<gg, 128, 0, stream>>>(x, Wq, Q,  mask, 1); // phi
  gemm_phi_mask<<<gg, 128, 0, stream>>>(x, Wk, Kb, mask, 3); // phi + mask
  gemm_phi_mask<<<gg, 128, 0, stream>>>(x, Wv, Vb, mask, 2); // mask

  ksum_kernel<<<dim3(16, 16), 256, 0, stream>>>(Kb, ksum);
  kv_accum<<<dim3(64, 8), 128, 0, stream>>>(Kb, Vb, kv);
  norm_kernel<<<dim3((BTROWS * NH) / 256), 256, 0, stream>>>(Q, ksum, nrm);
  qkv_apply<<<dim3(BTROWS / 128, NH), 128, 0, stream>>>(Q, kv, nrm);

  gemm_phi_mask<<<gg, 128, 0, stream>>>(Q, Wo, out, mask, 0); // final proj
}